// LinearInFieldChargesBlock_13872744366314
// MI455X (gfx1250) — compile-verified
//
#include <hip/hip_runtime.h>

typedef __attribute__((ext_vector_type(2))) float v2f;
typedef __attribute__((ext_vector_type(4))) float v4f;
typedef __attribute__((ext_vector_type(8))) float v8f;

#define KDIM        128
#define LSTRIDE     132                     // 128 + 4 pad -> conflict-free b64 LDS ops
#define NODES_PER_WG 64
#define WOFF        0
#define WSZ         (2 * 32 * LSTRIDE)      // 8448 floats: WT[s][m][k]
#define FOFF        WSZ
#define FSZ         (4 * NODES_PER_WG * LSTRIDE) // 33792 floats: F[a][n][k]
#define TOFF        FOFF                    // T(n,a,c) reuses F region after barrier
#define TSTRIDE     136                     // per-node stride (8B-aligned b64 stores)
#define ASTRIDE     34                      // per-a stride within a node row
#define SMEM_FLOATS (WSZ + FSZ)             // 42240
#define SMEM_BYTES  (SMEM_FLOATS * 4)       // 168960 B < 320 KB WGP LDS

__device__ __forceinline__ v8f wmma_f32_k4(v2f a, v2f b, v8f c) {
    // V_WMMA_F32_16X16X4_F32: D(16x16 f32) = A(16x4 f32) x B(4x16 f32) + C
    return __builtin_amdgcn_wmma_f32_16x16x4_f32(
        /*neg_a=*/false, a, /*neg_b=*/false, b,
        /*c_mod=*/(short)0, c, /*reuse_a=*/false, /*reuse_b=*/false);
}

extern "C" __global__ __launch_bounds__(256)
void fctp_wmma_kernel(const float* __restrict__ node_feat,
                      const float* __restrict__ pot_feat,
                      const float* __restrict__ w000,
                      const float* __restrict__ w011,
                      const float* __restrict__ w101,
                      const float* __restrict__ w110,
                      float* __restrict__ out,
                      int n_nodes)
{
    extern __shared__ float smem[];
    const int tid   = threadIdx.x;
    const int gbase = blockIdx.x * NODES_PER_WG;

    // ---- stage weights, transposed: WT(s,m,k) = W_s[k*16 + (m&15)] ----
    // s=0: [W000 | W011] (used with x0); s=1: [W110 | W101] (used with x1_i)
    for (int f = tid; f < 2 * 32 * KDIM; f += 256) {
        const int s = f >> 12;
        const int m = (f >> 7) & 31;
        const int k = f & 127;
        const float* wsrc = (s == 0) ? (m < 16 ? w000 : w011)
                                     : (m < 16 ? w110 : w101);
        smem[WOFF + (s * 32 + m) * LSTRIDE + k] = wsrc[k * 16 + (m & 15)];
    }

    // ---- stage node features, de-strided: F(a,n,k) ----
    // a=0: x0[k]; a=1+i: x1[k,i] (global layout 128 + 3k + i)
    for (int f = tid; f < NODES_PER_WG * KDIM; f += 256) { // float4 granules
        const int n = f >> 7;
        const int r = f & 127;
        if (gbase + n < n_nodes) {
            v4f d = *(const v4f*)(node_feat + (size_t)(gbase + n) * 512 + r * 4);
#pragma unroll
            for (int c = 0; c < 4; ++c) {
                const int e = r * 4 + c;
                int a, k;
                if (e < 128) { a = 0; k = e; }
                else         { const int e2 = e - 128; k = e2 / 3; a = 1 + (e2 - 3 * k); }
                smem[FOFF + (a * NODES_PER_WG + n) * LSTRIDE + k] = d[c];
            }
        }
    }
    __syncthreads();

    // ---- WMMA main loop: t(n,a,c) = sum_k F(a,n,k) * WT(s(a),c,k) ----
    // Wave = (node-group g, m-tile t): 4 accumulator chains (a=0..3), one tile.
    // Per K-step: 2 A loads (WA tile for a=0; WB tile shared by a=1..3) + 4 B loads
    // feed 4 WMMAs; loads are double-buffered so dscnt waits overlap WMMA issue.
    const int lane = tid & 31;
    const int wave = tid >> 5;
    const int g    = wave >> 1;  // node group (16 nodes on WMMA N dim)
    const int t    = wave & 1;   // output-column tile (m-base = 16*t)
    const int lm   = lane & 15;  // M / N index within tile
    const int lh   = lane >> 4;  // K half select (K = 2*lh + vgpr)

    const int aOff0 = WOFF + (     t * 16 + lm) * LSTRIDE + 2 * lh; // WA tile (a=0)
    const int aOff1 = WOFF + (32 + t * 16 + lm) * LSTRIDE + 2 * lh; // WB tile (a=1..3)
    int bOff[4];
#pragma unroll
    for (int a = 0; a < 4; ++a)
        bOff[a] = FOFF + (a * NODES_PER_WG + g * 16 + lm) * LSTRIDE + 2 * lh;

    v8f acc0 = {}, acc1 = {}, acc2 = {}, acc3 = {};

    v2f A0 = *(const v2f*)&smem[aOff0];
    v2f A1 = *(const v2f*)&smem[aOff1];
    v2f B0 = *(const v2f*)&smem[bOff[0]];
    v2f B1 = *(const v2f*)&smem[bOff[1]];
    v2f B2 = *(const v2f*)&smem[bOff[2]];
    v2f B3 = *(const v2f*)&smem[bOff[3]];

#pragma unroll 4
    for (int kc = 0; kc < KDIM; kc += 4) {
        const int nk = (kc + 4 < KDIM) ? (kc + 4) : 0; // wrap: harmless dummy prefetch
        // prefetch next chunk's operands first -> WMMAs below run on completed data
        const v2f nA0 = *(const v2f*)&smem[aOff0 + nk];
        const v2f nA1 = *(const v2f*)&smem[aOff1 + nk];
        const v2f nB0 = *(const v2f*)&smem[bOff[0] + nk];
        const v2f nB1 = *(const v2f*)&smem[bOff[1] + nk];
        const v2f nB2 = *(const v2f*)&smem[bOff[2] + nk];
        const v2f nB3 = *(const v2f*)&smem[bOff[3] + nk];

        acc0 = wmma_f32_k4(A0, B0, acc0);
        acc1 = wmma_f32_k4(A1, B1, acc1);
        acc2 = wmma_f32_k4(A1, B2, acc2);
        acc3 = wmma_f32_k4(A1, B3, acc3);

        A0 = nA0; A1 = nA1; B0 = nB0; B1 = nB1; B2 = nB2; B3 = nB3;
    }

    __syncthreads(); // everyone done reading F before T overwrites it

    // ---- scatter accumulators: T(n,a,c) at TOFF + n*TSTRIDE + a*ASTRIDE + c ----
    // lane holds c = 16*t + r + 8*lh for r=0..7 (consecutive) -> b64 stores
    {
        const int row = TOFF + (g * 16 + lm) * TSTRIDE + t * 16 + 8 * lh;
#pragma unroll
        for (int r2 = 0; r2 < 4; ++r2) {
            v2f p0 = { acc0[2 * r2], acc0[2 * r2 + 1] };
            v2f p1 = { acc1[2 * r2], acc1[2 * r2 + 1] };
            v2f p2 = { acc2[2 * r2], acc2[2 * r2 + 1] };
            v2f p3 = { acc3[2 * r2], acc3[2 * r2 + 1] };
            *(v2f*)&smem[row + 0 * ASTRIDE + 2 * r2] = p0;
            *(v2f*)&smem[row + 1 * ASTRIDE + 2 * r2] = p1;
            *(v2f*)&smem[row + 2 * ASTRIDE + 2 * r2] = p2;
            *(v2f*)&smem[row + 3 * ASTRIDE + 2 * r2] = p3;
        }
    }
    __syncthreads();

    // ---- epilogue: combine with pot features, scale, zero column 0 ----
    const int n = tid >> 2;   // 0..63
    const int q = tid & 3;    // q=0: out0[0..3]; q=1..3: out1[w=q-1][k=0..2]
    if (gbase + n < n_nodes) {
        const float C000 = 0.03125f;
        const float C110 = 0.018042195912175807f; // 1/(32*sqrt(3))
        const float C011 = 0.03125f;
        const float C101 = 0.03125f;
        const float* p  = pot_feat + (size_t)(gbase + n) * 16; // y0[v]=p[v]; y1[v,k]=p[4+3v+k]
        float*       o  = out      + (size_t)(gbase + n) * 16;
        const float* Tn = &smem[TOFF + n * TSTRIDE];
        if (q == 0) {
#pragma unroll
            for (int w = 0; w < 4; ++w) {
                float s = 0.f;
#pragma unroll
                for (int v = 0; v < 4; ++v) {
                    s += C000 * p[v] * Tn[0 * ASTRIDE + 4 * v + w];
#pragma unroll
                    for (int i = 0; i < 3; ++i)
                        s += C110 * p[4 + 3 * v + i] * Tn[(1 + i) * ASTRIDE + 4 * v + w];
                }
                o[w] = (w == 0) ? 0.0f : 0.01f * s;
            }
        } else {
            const int w = q - 1;
#pragma unroll
            for (int k = 0; k < 3; ++k) {
                float s = 0.f;
#pragma unroll
                for (int v = 0; v < 4; ++v) {
                    s += C011 * p[4 + 3 * v + k] * Tn[0 * ASTRIDE + 16 + 4 * v + w];
                    s += C101 * p[v]             * Tn[(1 + k) * ASTRIDE + 16 + 4 * v + w];
                }
                o[4 + 3 * w + k] = 0.01f * s;
            }
        }
    }
}

extern "C" void kernel_launch(void* const* d_in, const int* in_sizes, int n_in,
                              void* d_out, int out_size, void* d_ws, size_t ws_size,
                              hipStream_t stream) {
    const float* node_feat = (const float*)d_in[0];
    const float* pot_feat  = (const float*)d_in[1];
    const float* w000      = (const float*)d_in[2];
    const float* w011      = (const float*)d_in[3];
    const float* w101      = (const float*)d_in[4];
    const float* w110      = (const float*)d_in[5];
    float* out = (float*)d_out;

    const int n_nodes = in_sizes[0] / 512;                           // 400000
    const int grid    = (n_nodes + NODES_PER_WG - 1) / NODES_PER_WG; // 6250 exact

    fctp_wmma_kernel<<<grid, 256, SMEM_BYTES, stream>>>(
        node_feat, pot_feat, w000, w011, w101, w110, out, n_nodes);
}